// GCN_31198642438704
// MI455X (gfx1250) — compile-verified
//
#include <hip/hip_runtime.h>
#include <math.h>

// Problem constants from the reference
#define BATCH 256
#define NNODE 512
#define MAXD  6
#define NDEG  7
#define FIN   62
#define HD    128
#define BN    (BATCH * NNODE)   // 131072 nodes total

typedef __attribute__((ext_vector_type(2))) float v2f;
typedef __attribute__((ext_vector_type(8))) float v8f;

// ---------------------------------------------------------------------------
// NFP conv, fused: (self + neighbor-sum) staged in LDS, then per-degree
// fp32 WMMA (V_WMMA_F32_16X16X4_F32) with per-row degree selection.
//   X : [BN, F]  input features (F=62 layer1, F=128 layer2)
//   E : [BN, 6]  neighbor indices within batch, -1 = padding
//   W : [7, F, 128], Bv : [7, 128]
//   O : [BN, 128]
// Block = 256 threads (8 wave32s). Block owns 64 nodes (4 M-tiles of 16).
// Wave w owns output columns [16w, 16w+16). B tile loaded once per (d,k)
// and reused across the 4 M-tiles.
// ---------------------------------------------------------------------------
template <int KPAD, int F>
__global__ __launch_bounds__(256) void nfp_conv_wmma(
    const float* __restrict__ X, const int* __restrict__ E,
    const float* __restrict__ W, const float* __restrict__ Bv,
    float* __restrict__ O)
{
    __shared__ float sA[64][KPAD + 1];   // +1 pad: consecutive rows -> distinct banks
    __shared__ int   sdeg[64];

    const int tid  = threadIdx.x;
    const int base = blockIdx.x * 64;    // first flat node of this block

    // -------- Phase 1: summed = self + sum(valid neighbors), zero-padded ----
    for (int idx = tid; idx < 64 * KPAD; idx += 256) {
        const int m = idx / KPAD;
        const int k = idx % KPAD;
        const int node = base + m;
        const int bb   = node >> 9;      // node / 512
        float v = 0.0f;
        if (k < F) {
            v = X[(size_t)node * F + k];
#pragma unroll
            for (int j = 0; j < MAXD; ++j) {
                const int e  = E[node * MAXD + j];
                const int nb = (bb << 9) + (e < 0 ? 0 : e);
                const float xv = X[(size_t)nb * F + k];
                v += (e < 0) ? 0.0f : xv;
            }
        }
        sA[m][k] = v;
    }
    if (tid < 64) {
        const int node = base + tid;
        int dg = 0;
#pragma unroll
        for (int j = 0; j < MAXD; ++j) dg += (E[node * MAXD + j] >= 0) ? 1 : 0;
        sdeg[tid] = dg;
    }
    __syncthreads();

    // -------- Phase 2: WMMA over 7 degree buckets, select per row ----------
    const int lane   = tid & 31;
    const int wave   = tid >> 5;
    const int hi     = lane >> 4;            // half-wave select
    const int mrow   = lane & 15;            // A-matrix row within M-tile
    const int ncol   = (wave << 4) + (lane & 15);  // output column 0..127
    const int rowoff = hi << 3;              // C/D rows: M = vgpr + 8*hi

    const v8f vzero = {0.f, 0.f, 0.f, 0.f, 0.f, 0.f, 0.f, 0.f};
    v8f res[4];
#pragma unroll
    for (int t = 0; t < 4; ++t) res[t] = vzero;

    for (int d = 0; d < NDEG; ++d) {
        v8f acc[4];
#pragma unroll
        for (int t = 0; t < 4; ++t) acc[t] = vzero;

        for (int k0 = 0; k0 < KPAD; k0 += 4) {
            const int kk = k0 + (hi << 1);   // lanes 16-31 hold K = k0+2,k0+3
            // B tile (4x16, K-major per ISA layout); guard fp32 pad cols
            v2f bv;
            bv.x = (kk     < F) ? W[((size_t)d * F + kk    ) * HD + ncol] : 0.0f;
            bv.y = (kk + 1 < F) ? W[((size_t)d * F + kk + 1) * HD + ncol] : 0.0f;
#pragma unroll
            for (int t = 0; t < 4; ++t) {
                v2f av;                       // A tile 16x4, fp32 layout
                av.x = sA[t * 16 + mrow][kk];
                av.y = sA[t * 16 + mrow][kk + 1];
                acc[t] = __builtin_amdgcn_wmma_f32_16x16x4_f32(
                    /*neg_a=*/false, av, /*neg_b=*/false, bv,
                    /*c_mod=*/(short)0, acc[t],
                    /*reuse_a=*/false, /*reuse_b=*/false);
            }
        }

        const float bias = Bv[d * HD + ncol];
#pragma unroll
        for (int t = 0; t < 4; ++t) {
#pragma unroll
            for (int i = 0; i < 8; ++i) {
                const int m = t * 16 + i + rowoff;
                const float h = 1.0f / (1.0f + __expf(-(acc[t][i] + bias)));
                res[t][i] += (sdeg[m] == d) ? h : 0.0f;   // exactly one d matches
            }
        }
    }

    // -------- Store (C/D layout: row = vgpr + 8*hi, col = lane&15) ---------
#pragma unroll
    for (int t = 0; t < 4; ++t)
#pragma unroll
        for (int i = 0; i < 8; ++i) {
            const int m = t * 16 + i + rowoff;
            O[(size_t)(base + m) * HD + ncol] = res[t][i];
        }
}

// ---------------------------------------------------------------------------
// graph_pool: per (node, feature) max over self + valid neighbors.
// ---------------------------------------------------------------------------
__global__ __launch_bounds__(256) void pool_max(
    const float* __restrict__ X, const int* __restrict__ E,
    float* __restrict__ Y)
{
    const size_t idx = (size_t)blockIdx.x * 256 + threadIdx.x;  // over BN*HD
    const int f    = (int)(idx & (HD - 1));
    const int node = (int)(idx >> 7);
    const int bb   = node >> 9;
    float v = X[(size_t)node * HD + f];
#pragma unroll
    for (int j = 0; j < MAXD; ++j) {
        const int e  = E[node * MAXD + j];
        const int nb = (bb << 9) + (e < 0 ? 0 : e);
        const float xv = X[(size_t)nb * HD + f];
        v = (e >= 0) ? fmaxf(v, xv) : v;
    }
    Y[idx] = v;
}

// ---------------------------------------------------------------------------
// SubgraphSum: out[b,h] = sum_n Y[b,n,h]
// ---------------------------------------------------------------------------
__global__ __launch_bounds__(256) void reduce_sum(
    const float* __restrict__ Y, float* __restrict__ O)
{
    const int idx = blockIdx.x * 256 + threadIdx.x;   // over BATCH*HD
    const int h  = idx & (HD - 1);
    const int bb = idx >> 7;
    const float* p = Y + (size_t)bb * NNODE * HD + h;
    float s = 0.0f;
    for (int nn = 0; nn < NNODE; ++nn) s += p[(size_t)nn * HD];
    O[idx] = s;
}

// ---------------------------------------------------------------------------
extern "C" void kernel_launch(void* const* d_in, const int* in_sizes, int n_in,
                              void* d_out, int out_size, void* d_ws, size_t ws_size,
                              hipStream_t stream)
{
    const float* a  = (const float*)d_in[0];   // [B,N,62]
    const int*   e  = (const int*)  d_in[1];   // [B,N,6]
    const float* W0 = (const float*)d_in[2];   // [7,62,128]
    const float* b0 = (const float*)d_in[3];   // [7,128]
    const float* W1 = (const float*)d_in[4];   // [7,128,128]
    const float* b1 = (const float*)d_in[5];   // [7,128]
    float* out = (float*)d_out;                // [B,128]

    float* bufA = (float*)d_ws;                  // [BN,128]  x1 / x2
    float* bufB = bufA + (size_t)BN * HD;        // [BN,128]  y1 / y2

    dim3 blk(256);
    // layer 1: conv (62->128, K padded to 64) then pool
    nfp_conv_wmma<64, FIN><<<dim3(BN / 64), blk, 0, stream>>>(a, e, W0, b0, bufA);
    pool_max<<<dim3((BN * HD) / 256), blk, 0, stream>>>(bufA, e, bufB);
    // layer 2: conv (128->128) then pool
    nfp_conv_wmma<128, HD><<<dim3(BN / 64), blk, 0, stream>>>(bufB, e, W1, b1, bufA);
    pool_max<<<dim3((BN * HD) / 256), blk, 0, stream>>>(bufA, e, bufB);
    // sum over nodes
    reduce_sum<<<dim3((BATCH * HD) / 256), blk, 0, stream>>>(bufB, out);
}